// LengthRegulator_87591563035283
// MI455X (gfx1250) — compile-verified
//
#include <hip/hip_runtime.h>
#include <stdint.h>

#define BB 32
#define TT 512
#define DD 384
#define MAX_FRAMES 2048
#define CHUNKS 64
#define FPB (MAX_FRAMES / CHUNKS)   // 32 frames per block
#define FPW (FPB / 8)               // 4 frames per wave (8 waves of 32)

typedef int v4i_t __attribute__((vector_size(16)));             // int4 vector
typedef v4i_t __attribute__((address_space(3))) as3_v4i;        // LDS int4

// ---------------------------------------------------------------------------
// Kernel 1: per-batch inclusive scan of durations (with all-zero-row fix).
// One block per batch, 512 threads, Hillis-Steele scan in LDS.
// ---------------------------------------------------------------------------
__global__ __launch_bounds__(TT) void lr_scan_kernel(const int* __restrict__ ds,
                                                     int* __restrict__ cum) {
    const int b   = blockIdx.x;
    const int tid = threadIdx.x;
    __shared__ int s[TT];

    s[tid] = ds[b * TT + tid];
    __syncthreads();

    #pragma unroll
    for (int off = 1; off < TT; off <<= 1) {
        int t = (tid >= off) ? s[tid - off] : 0;
        __syncthreads();
        s[tid] += t;
        __syncthreads();
    }

    const int total = s[TT - 1];
    // row_all_zero -> ds := 1 everywhere -> cum[t] = t+1
    cum[b * TT + tid] = (total == 0) ? (tid + 1) : s[tid];
}

// ---------------------------------------------------------------------------
// Kernel 2: frame expansion (gather). One block = one batch x 32 frames.
// cum[b] table staged to LDS via async-to-LDS DMA when available; 32 threads
// binary-search LDS; each wave32 copies 4 rows with float4 (b128) traffic.
// ---------------------------------------------------------------------------
__global__ __launch_bounds__(256) void lr_expand_kernel(const float* __restrict__ xs,
                                                        const int*   __restrict__ cum,
                                                        float*       __restrict__ out) {
    const int b     = blockIdx.y;
    const int chunk = blockIdx.x;
    const int tid   = threadIdx.x;

    __shared__ int s_cum[TT];          // 2 KB cumsum table
    __shared__ int s_idx[FPB];
    __shared__ int s_val[FPB];

    const int* gcum = cum + b * TT;

    // ---- stage cum[b][0..511] into LDS -----------------------------------
#if defined(__gfx1250__) && __has_builtin(__builtin_amdgcn_global_load_async_to_lds_b128)
    if (tid < TT / 4) {  // 128 threads x 16B = 2KB
        __builtin_amdgcn_global_load_async_to_lds_b128(
            (v4i_t*)(gcum + tid * 4),            // global src (C-cast drops const)
            (as3_v4i*)(&s_cum[tid * 4]),         // LDS dst (addrspace(3) int4*)
            /*offset=*/0, /*cpol=*/0);
    }
  #if __has_builtin(__builtin_amdgcn_s_wait_asynccnt)
    __builtin_amdgcn_s_wait_asynccnt(0);
  #else
    asm volatile("s_wait_asynccnt 0" ::: "memory");
  #endif
#else
    for (int i = tid; i < TT; i += 256) s_cum[i] = gcum[i];
#endif
    __syncthreads();

    // ---- binary search: frame t -> source token index --------------------
    if (tid < FPB) {
        const int t  = chunk * FPB + tid;
        int lo = 0, hi = TT;
        while (lo < hi) {                    // 9 ds_load steps, LDS-resident
            int mid = (lo + hi) >> 1;
            if (s_cum[mid] <= t) lo = mid + 1; else hi = mid;
        }
        s_idx[tid] = (lo < TT) ? lo : (TT - 1);
        s_val[tid] = (t < s_cum[TT - 1]) ? 1 : 0;
    }
    __syncthreads();

    // ---- row copy: one wave per frame, lane moves 3 x float4 -------------
    const int wave = tid >> 5;
    const int lane = tid & 31;

    // warm L2 with this wave's source rows (global_prefetch_b8 path)
    #pragma unroll
    for (int i = 0; i < FPW; ++i) {
        const int f = wave * FPW + i;
        if (s_val[f])
            __builtin_prefetch(xs + ((size_t)b * TT + s_idx[f]) * DD + lane * 12, 0, 3);
    }

    #pragma unroll
    for (int i = 0; i < FPW; ++i) {
        const int f   = wave * FPW + i;
        const int t   = chunk * FPB + f;
        float4* dst = (float4*)(out + ((size_t)b * MAX_FRAMES + t) * DD);
        if (s_val[f]) {
            const float4* src =
                (const float4*)(xs + ((size_t)b * TT + s_idx[f]) * DD);
            #pragma unroll
            for (int k = 0; k < 3; ++k) {
                const int c4 = k * 32 + lane;    // 96 float4 per row
                dst[c4] = src[c4];
            }
        } else {
            const float4 z = make_float4(0.f, 0.f, 0.f, 0.f);
            #pragma unroll
            for (int k = 0; k < 3; ++k) dst[k * 32 + lane] = z;
        }
    }
}

// ---------------------------------------------------------------------------
extern "C" void kernel_launch(void* const* d_in, const int* in_sizes, int n_in,
                              void* d_out, int out_size, void* d_ws, size_t ws_size,
                              hipStream_t stream) {
    const float* xs  = (const float*)d_in[0];
    const int*   ds  = (const int*)d_in[1];
    float*       out = (float*)d_out;
    int*         cum = (int*)d_ws;          // B*T ints = 64 KB scratch

    lr_scan_kernel<<<BB, TT, 0, stream>>>(ds, cum);
    lr_expand_kernel<<<dim3(CHUNKS, BB), 256, 0, stream>>>(xs, cum, out);
}